// MultiHeadAttentionQuantum_65481071399413
// MI455X (gfx1250) — compile-verified
//
#include <hip/hip_runtime.h>
#include <hip/hip_bf16.h>

typedef __attribute__((ext_vector_type(2))) float v2f;
typedef __attribute__((ext_vector_type(8))) float v8f;

#define NB 8
#define NS 512
#define NE 64
#define NH 8
#define NW 8

__device__ __forceinline__ v8f wmma4(v2f a, v2f b, v8f c) {
    // D = A(16x4 f32) * B(4x16 f32) + C(16x16 f32)
    return __builtin_amdgcn_wmma_f32_16x16x4_f32(
        /*neg_a=*/false, a, /*neg_b=*/false, b,
        /*c_mod=*/(short)0, c, /*reuse_a=*/false, /*reuse_b=*/false);
}

// ---------------------------------------------------------------------------
// Stage 1: the 8-qubit circuit collapses to cosine prefix products.
//   exp[0]   = prod_{j=1..7} cos(x_j + theta_j)
//   exp[w>0] = prod_{j=0..w} cos(x_j + theta_j)
// qbuf layout: [B, H, S, 8]
// ---------------------------------------------------------------------------
__global__ void qheads_kernel(const float* __restrict__ x,
                              const float* __restrict__ theta,
                              float* __restrict__ qbuf) {
    int t = blockIdx.x * blockDim.x + threadIdx.x;   // one thread per (b,s,h)
    int b   = t >> 12;            // / (S*H) = 4096
    int rem = t & 4095;
    int s   = rem >> 3;
    int h   = rem & 7;

    const float* xp = x + ((b * NS + s) * NE + h * NW);
    float u[NW];
#pragma unroll
    for (int w = 0; w < NW; ++w) u[w] = __cosf(xp[w] + theta[w]);

    float e[NW];
    float p = u[1];
#pragma unroll
    for (int w = 2; w < NW; ++w) p *= u[w];
    e[0] = p;
    float run = u[0];
#pragma unroll
    for (int w = 1; w < NW; ++w) { run *= u[w]; e[w] = run; }

    float* qp = qbuf + (((b * NH + h) * NS) + s) * NW;
#pragma unroll
    for (int w = 0; w < NW; ++w) qp[w] = e[w];
}

// ---------------------------------------------------------------------------
// Stage 2: streaming softmax attention, one wave per (b,h, 16-row query tile).
// Whole K/V array for this (b,h) (16 KB) staged into LDS once; the 32-tile
// loop is pure LDS + WMMA + v_exp. Scores bounded by 8/sqrt(8): no max
// subtraction needed. Softmax computed in base 2 with log2(e)/sqrt(8) folded
// into the Q fragment (v_exp_f32 is natively 2^x; base cancels in the ratio).
// Denominator rides in column 8 of the ctx WMMA via a ones-column in B.
// ctxbuf layout: [B, S, E] (inner index h*8 + col)
// ---------------------------------------------------------------------------
__global__ __launch_bounds__(32)
void qattn_kernel(const float* __restrict__ qbuf, float* __restrict__ ctxbuf) {
    const int lane  = threadIdx.x;
    const int bh    = blockIdx.x >> 5;     // 64 (b,h) pairs
    const int mtile = blockIdx.x & 31;     // 32 query tiles of 16
    const int r     = lane & 15;           // row (A) / col (B,D) index
    const int o     = (lane >> 4) * 2;     // K sub-offset for upper lane half
    const float scale = 0.51012258641038f; // log2(e) / sqrt(8)

    __shared__ float kt[NS * NW];   // 16 KB: all keys/values for this (b,h)
    __shared__ float et[16 * 16];   // 2^score tile for D->A relayout

    // cooperative load of the full K array: 4096 floats = 32 float4 per lane
    const float* qbase = qbuf + bh * (NS * NW);
#pragma unroll 4
    for (int j = 0; j < 32; ++j) {
        int idx = (j * 32 + lane) * 4;
        *(float4*)&kt[idx] = *(const float4*)(qbase + idx);
    }
    __syncthreads();

    // A fragments: Q rows mtile*16 + r, pre-scaled by log2(e)/sqrt(8)
    const float* qrow = &kt[(mtile * 16 + r) * NW];
    v2f a0, a4;
    a0.x = qrow[0 + o] * scale; a0.y = qrow[1 + o] * scale;
    a4.x = qrow[4 + o] * scale; a4.y = qrow[5 + o] * scale;

    const float one8  = (r == 8) ? 1.0f : 0.0f;   // ones column -> denominator
    const int   rc    = r & 7;                    // clamped feature index
    const bool  rlo   = (r < 8);

    v8f acc = {0.f, 0.f, 0.f, 0.f, 0.f, 0.f, 0.f, 0.f};
    const v8f zero = {0.f, 0.f, 0.f, 0.f, 0.f, 0.f, 0.f, 0.f};

    for (int ktile = 0; ktile < 32; ++ktile) {
        const int base = ktile * 16 * NW;

        // scores = Q K^T : B[k][n] = K[n][k]
        v2f b0, b4;
        b0.x = kt[base + r * 8 + 0 + o]; b0.y = kt[base + r * 8 + 1 + o];
        b4.x = kt[base + r * 8 + 4 + o]; b4.y = kt[base + r * 8 + 5 + o];
        v8f s = wmma4(a0, b0, zero);
        s = wmma4(a4, b4, s);

        // 2^x directly on raw WMMA output (scale pre-folded into A)
        float e[8];
#pragma unroll
        for (int i = 0; i < 8; ++i) e[i] = __builtin_amdgcn_exp2f(s[i]);

        __syncthreads();  // previous iteration's et readers are done
#pragma unroll
        for (int i = 0; i < 8; ++i)
            et[(i + ((lane >> 4) << 3)) * 16 + r] = e[i];
        __syncthreads();  // et visible

        // ctx += E (16x16) @ [V | 1 | 0] (16x16)
#pragma unroll
        for (int k0 = 0; k0 < 16; k0 += 4) {
            v2f ea, eb;
            ea.x = et[r * 16 + k0 + o];
            ea.y = et[r * 16 + k0 + o + 1];
            // unconditional clamped loads, then cndmask selects (EXEC stays full)
            float kv0 = kt[base + (k0 + o)     * 8 + rc];
            float kv1 = kt[base + (k0 + o + 1) * 8 + rc];
            eb.x = rlo ? kv0 : one8;
            eb.y = rlo ? kv1 : one8;
            acc = wmma4(ea, eb, acc);
        }
    }

    // normalize: denominator in col 8 (lane 8 rows 0-7, lane 24 rows 8-15)
#pragma unroll
    for (int i = 0; i < 8; ++i) {
        float den = __shfl(acc[i], (lane & 16) + 8, 32);
        acc[i] = acc[i] / den;
    }

    // store cols 0..7 of each row into [B,S,E] with inner index h*8+col
    if (rlo) {
        int b = bh >> 3, h = bh & 7;
#pragma unroll
        for (int i = 0; i < 8; ++i) {
            int row = mtile * 16 + i + ((lane >> 4) << 3);
            ctxbuf[((b * NS + row) * NE) + h * NW + r] = acc[i];
        }
    }
}

// ---------------------------------------------------------------------------
// Stage 3: out = ctx @ W_o^T + b_o.  M=4096, K=64, N=64.
// One wave per 16-row strip; 4 N-tiles share the A fragment. 64 WMMAs/wave.
// ---------------------------------------------------------------------------
__global__ __launch_bounds__(32)
void proj_kernel(const float* __restrict__ ctxbuf,
                 const float* __restrict__ Wo,
                 const float* __restrict__ bo,
                 float* __restrict__ out) {
    const int lane  = threadIdx.x;
    const int r     = lane & 15;
    const int o     = (lane >> 4) * 2;
    const int mtile = blockIdx.x;     // 0..255

    const float* arow = ctxbuf + (mtile * 16 + r) * NE;

    v8f acc[4];
#pragma unroll
    for (int nt = 0; nt < 4; ++nt)
        acc[nt] = (v8f){0.f, 0.f, 0.f, 0.f, 0.f, 0.f, 0.f, 0.f};

#pragma unroll 4
    for (int k0 = 0; k0 < NE; k0 += 4) {
        v2f a;
        a.x = arow[k0 + o]; a.y = arow[k0 + o + 1];
#pragma unroll
        for (int nt = 0; nt < 4; ++nt) {
            // out[m][n] = sum_k ctx[m][k] * W_o[n][k]  ->  B[k][n] = W_o[n][k]
            v2f b;
            b.x = Wo[(nt * 16 + r) * NE + k0 + o];
            b.y = Wo[(nt * 16 + r) * NE + k0 + o + 1];
            acc[nt] = wmma4(a, b, acc[nt]);
        }
    }

#pragma unroll
    for (int nt = 0; nt < 4; ++nt) {
        float bias = bo[nt * 16 + r];
#pragma unroll
        for (int i = 0; i < 8; ++i) {
            int row = mtile * 16 + i + ((lane >> 4) << 3);
            out[row * NE + nt * 16 + r] = acc[nt][i] + bias;
        }
    }
}

extern "C" void kernel_launch(void* const* d_in, const int* in_sizes, int n_in,
                              void* d_out, int out_size, void* d_ws, size_t ws_size,
                              hipStream_t stream) {
    const float* x     = (const float*)d_in[0];   // [8,512,64]
    const float* theta = (const float*)d_in[1];   // [8]
    const float* Wo    = (const float*)d_in[2];   // [64,64]
    const float* bo    = (const float*)d_in[3];   // [64]
    float* out = (float*)d_out;                   // [8,512,64]

    float* qbuf   = (float*)d_ws;                 // [8,8,512,8]  = 262144 f32
    float* ctxbuf = qbuf + NB * NH * NS * NW;     // [8,512,64]   = 262144 f32

    qheads_kernel<<<(NB * NS * NH) / 256, 256, 0, stream>>>(x, theta, qbuf);
    qattn_kernel<<<NB * NH * (NS / 16), 32, 0, stream>>>(qbuf, ctxbuf);
    proj_kernel<<<(NB * NS) / 16, 32, 0, stream>>>(ctxbuf, Wo, bo, out);
}